// GatLayer_7980049236118
// MI455X (gfx1250) — compile-verified
//
#include <hip/hip_runtime.h>
#include <hip/hip_bf16.h>

// ---------------------------------------------------------------------------
// GAT layer for MI455X (gfx1250, wave32).
//   h = X @ W                      -> V_WMMA_F32_16X16X4_F32 (fp32-exact)
//   s1[n] = h[n] . a[0:64]
//   s2[n] = h[n] . a[64:128]
//   alpha_e = exp(leakyrelu(s1[src]+s2[dst]))   (edges + N self loops)
//   denom[src] += alpha_e          (global f32 atomics, L2-resident)
//   out[src]  += (alpha_e/denom[src]) * h[dst]  (wave-per-edge scatter)
// Memory-bound: ~870 MB of L2-resident gather/scatter dominates; FLOPs trivial.
// ---------------------------------------------------------------------------

typedef __attribute__((ext_vector_type(2))) float v2f;
typedef __attribute__((ext_vector_type(8))) float v8f;

#define SLOPE 0.2f
#define F 64

// ---------------------------------------------------------------- zero fill
__global__ void gat_zero(float* __restrict__ p, int n) {
    int i = blockIdx.x * blockDim.x + threadIdx.x;
    if (i < n) p[i] = 0.0f;
}

// ------------------------------------------------------- h = X @ W via WMMA
// Block = 256 threads = 8 waves. Each wave computes one 16x16 tile of H.
// Wave w: row tile = w/4 (2 per block -> 32 rows/block), col tile = w%4.
// A fragment (16x4 f32, ISA 7.12.2): lane = half*16 + M ; VGPR v -> K = half*2+v.
// B fragment (4x16 f32): lane = half*16 + Ncol ; VGPR v -> K = half*2+v.
// C/D (16x16 f32): lane = half*16 + Ncol ; VGPR r -> M = r + 8*half.
__global__ __launch_bounds__(256) void gat_gemm_wmma(
        const float* __restrict__ X,   // [N,64]
        const float* __restrict__ W,   // [64,64]
        float* __restrict__ H,         // [N,64]
        int N) {
    const int wave = threadIdx.x >> 5;
    const int lane = threadIdx.x & 31;
    const int row_base = blockIdx.x * 32 + (wave >> 2) * 16;   // wave-uniform
    const int col_base = (wave & 3) * 16;
    if (row_base >= N) return;   // wave-uniform: EXEC stays all-1s for WMMA

    const int half = lane >> 4;        // 0..1
    const int ml   = lane & 15;        // M (A) or N-col (B, C/D)

    const float* xr = X + (size_t)(row_base + ml) * F;   // row of A
    v8f c = {};
    #pragma unroll
    for (int k0 = 0; k0 < F; k0 += 4) {
        const int ka = k0 + half * 2;
        v2f a;
        a.x = xr[ka];
        a.y = xr[ka + 1];
        v2f b;
        b.x = W[(size_t)ka       * F + col_base + ml];
        b.y = W[(size_t)(ka + 1) * F + col_base + ml];
        // 8 args: (neg_a, A, neg_b, B, c_mod, C, reuse_a, reuse_b)
        c = __builtin_amdgcn_wmma_f32_16x16x4_f32(
                false, a, false, b, (short)0, c, false, false);
    }
    #pragma unroll
    for (int r = 0; r < 8; ++r) {
        H[(size_t)(row_base + r + half * 8) * F + col_base + ml] = c[r];
    }
}

// ----------------------------------------------- per-node attention scores
__global__ void gat_scores(const float* __restrict__ H,
                           const float* __restrict__ a,
                           float* __restrict__ s1,
                           float* __restrict__ s2, int N) {
    int n = blockIdx.x * blockDim.x + threadIdx.x;
    if (n >= N) return;
    const float4* hr = (const float4*)(H + (size_t)n * F);
    const float4* a1 = (const float4*)a;
    const float4* a2 = (const float4*)(a + F);
    float x = 0.0f, y = 0.0f;
    #pragma unroll
    for (int i = 0; i < F / 4; ++i) {
        float4 h4 = hr[i];
        float4 p  = a1[i];
        float4 q  = a2[i];
        x += h4.x * p.x + h4.y * p.y + h4.z * p.z + h4.w * p.w;
        y += h4.x * q.x + h4.y * q.y + h4.z * q.z + h4.w * q.w;
    }
    s1[n] = x;
    s2[n] = y;
}

__device__ __forceinline__ float edge_alpha(const float* s1, const float* s2,
                                            int s, int d) {
    float t = s1[s] + s2[d];
    t = (t >= 0.0f) ? t : SLOPE * t;
    return expf(t);
}

// -------------------------------------- pass 1: softmax denominators by src
// i in [0, E)      : real edge (src[i], dst[i])
// i in [E, E+N)    : self loop (n, n), n = i - E
__global__ void gat_denom(const int* __restrict__ src,
                          const int* __restrict__ dst,
                          const float* __restrict__ s1,
                          const float* __restrict__ s2,
                          float* __restrict__ denom, int E, int N) {
    int i = blockIdx.x * blockDim.x + threadIdx.x;
    if (i >= E + N) return;
    int s, d;
    if (i < E) { s = src[i]; d = dst[i]; }
    else       { s = d = i - E; }
    atomicAdd(&denom[s], edge_alpha(s1, s2, s, d));
}

// ----------------------- pass 2: out[src] += (alpha/denom[src]) * h[dst]
// One wave per edge, 2 features per lane (float2 gather + 2 f32 atomics).
__global__ __launch_bounds__(256) void gat_scatter(
        const int* __restrict__ src,
        const int* __restrict__ dst,
        const float* __restrict__ s1,
        const float* __restrict__ s2,
        const float* __restrict__ denom,
        const float* __restrict__ H,
        float* __restrict__ out, int E, int N) {
    const int lane = threadIdx.x & 31;
    const int i = blockIdx.x * 8 + (threadIdx.x >> 5);   // wave-uniform edge id
    if (i >= E + N) return;
    int s, d;
    if (i < E) { s = src[i]; d = dst[i]; }
    else       { s = d = i - E; }
    const float w = edge_alpha(s1, s2, s, d) / denom[s];
    const float2 v = ((const float2*)(H + (size_t)d * F))[lane];
    float* o = out + (size_t)s * F + lane * 2;
    atomicAdd(o,     v.x * w);
    atomicAdd(o + 1, v.y * w);
}

// ---------------------------------------------------------------------------
extern "C" void kernel_launch(void* const* d_in, const int* in_sizes, int n_in,
                              void* d_out, int out_size, void* d_ws, size_t ws_size,
                              hipStream_t stream) {
    const float* X     = (const float*)d_in[0];   // [N, 64]
    const int*   edges = (const int*)  d_in[1];   // [2, E] flattened
    const float* W     = (const float*)d_in[2];   // [64, 64]
    const float* a     = (const float*)d_in[3];   // [128]
    float*       out   = (float*)d_out;           // [N, 64]

    const int N = in_sizes[0] / F;     // 100000
    const int E = in_sizes[1] / 2;     // 1600000
    const int M = E + N;               // edges + self loops
    const int* src = edges;
    const int* dst = edges + E;

    // workspace layout (floats): H[N*64] | s1[N] | s2[N] | denom[N]
    float* H     = (float*)d_ws;
    float* s1    = H + (size_t)N * F;
    float* s2    = s1 + N;
    float* denom = s2 + N;

    // 0) zero accumulators (out and denom are atomically accumulated)
    gat_zero<<<(N * F + 255) / 256, 256, 0, stream>>>(out, N * F);
    gat_zero<<<(N + 255) / 256, 256, 0, stream>>>(denom, N);

    // 1) h = X @ W  (WMMA f32 16x16x4; 32 rows x 64 cols per block)
    gat_gemm_wmma<<<(N + 31) / 32, 256, 0, stream>>>(X, W, H, N);

    // 2) per-node scores s1, s2
    gat_scores<<<(N + 255) / 256, 256, 0, stream>>>(H, a, s1, s2, N);

    // 3) softmax denominators
    gat_denom<<<(M + 255) / 256, 256, 0, stream>>>(src, dst, s1, s2, denom, E, N);

    // 4) weighted scatter-add (one wave per edge)
    gat_scatter<<<(M + 7) / 8, 256, 0, stream>>>(src, dst, s1, s2, denom, H, out, E, N);
}